// ROIAlign_90649579749430
// MI455X (gfx1250) — compile-verified
//
#include <hip/hip_runtime.h>
#include <hip/hip_bf16.h>

#define PH_ 7
#define PW_ 7
#define SR_ 2
#define C_ 256
#define H_ 100
#define W_ 100
#define HW_ (H_ * W_)
#define SCALE_ 0.125f
#define NBIN_ (PH_ * PW_)          // 49
#define GH_ (PH_ * SR_)            // 14
#define OUTPERROI_ (C_ * NBIN_)    // 12544 floats per ROI

// ---------------------------------------------------------------------------
// Kernel 1: NCHW -> NHWC transpose of the feature map, staged through LDS
// using the CDNA5 async global->LDS data path (ASYNCcnt-tracked).
// Grid: (ceil(HW/32), C/32, B), block (32, 8).
// b32 asyncs kept on purpose: 33-float row pad is bank-conflict-free for the
// transposed reads; the whole stage moves only 40 MB once (~2us at HBM rate).
// ---------------------------------------------------------------------------
__global__ __launch_bounds__(256) void nchw_to_nhwc_async(
    const float* __restrict__ in, float* __restrict__ out) {
  __shared__ float tile[32][33];                     // +1 pad: conflict-free
  const int b   = blockIdx.z;
  const int c0  = blockIdx.y * 32;
  const int hw0 = blockIdx.x * 32;
  const int tx  = threadIdx.x;                       // 0..31
  const int ty  = threadIdx.y;                       // 0..7

  // Async-copy tile[c_local][hw_local] = in[b][c0+c_local][hw0+tx]
  const int hwL = min(hw0 + tx, HW_ - 1);            // clamp (value unused if OOB)
#pragma unroll
  for (int j = 0; j < 4; ++j) {
    const int cl = ty + 8 * j;
    const unsigned long long gaddr = (unsigned long long)(uintptr_t)(
        in + (size_t)(b * C_ + c0 + cl) * HW_ + hwL);
    const unsigned ldst = (unsigned)(uintptr_t)(&tile[cl][tx]);
    asm volatile("global_load_async_to_lds_b32 %0, %1, off"
                 :: "v"(ldst), "v"(gaddr) : "memory");
  }
  asm volatile("s_wait_asynccnt 0" ::: "memory");    // this wave's async ops done
  __syncthreads();                                   // cross-wave visibility

  // Coalesced store: out[b][hw][c], consecutive lanes -> consecutive c
#pragma unroll
  for (int j = 0; j < 4; ++j) {
    const int hwl = ty + 8 * j;
    const int hw  = hw0 + hwl;
    if (hw < HW_) {
      out[((size_t)b * HW_ + hw) * C_ + (c0 + tx)] = tile[tx][hwl];
    }
  }
}

// ---------------------------------------------------------------------------
// Kernel 2: ROIAlign gather. One block (256 threads) per ROI; lane == channel.
// Strides parameterize NHWC (coalesced) vs NCHW (fallback) layouts.
// Per-ROI sample metadata computed once into LDS; results staged in LDS
// [c][bin] (stride 49 floats: coprime with 64 banks -> conflict-free) and
// streamed out as one contiguous coalesced 49 KB float4 stream per ROI.
// ---------------------------------------------------------------------------
__global__ __launch_bounds__(256) void roi_align_kernel(
    const float* __restrict__ feat, const float* __restrict__ rois,
    float* __restrict__ out,
    long bStride, int yStride, int xStride, int cStride) {
  __shared__ int   iy0[GH_], iy1[GH_], ix0[GH_], ix1[GH_];
  __shared__ float fly[GH_], fvy[GH_], flx[GH_], fvx[GH_];
  __shared__ float sout[OUTPERROI_];                 // 50,176 B

  const int tid = threadIdx.x;                       // == channel
  const int k   = blockIdx.x;

  const float* __restrict__ r = rois + (size_t)k * 5;
  const int   b   = (int)r[0];
  const float rx1 = r[1] * SCALE_;
  const float ry1 = r[2] * SCALE_;
  const float rx2 = r[3] * SCALE_;
  const float ry2 = r[4] * SCALE_;
  const float roi_w = fmaxf(rx2 - rx1, 1.0f);
  const float roi_h = fmaxf(ry2 - ry1, 1.0f);
  const float step_x = roi_w * (1.0f / (PW_ * SR_)); // bin_w / SR
  const float step_y = roi_h * (1.0f / (PH_ * SR_)); // bin_h / SR

  // Lanes 0..13 compute the 14 y-sample records, lanes 16..29 the x-records.
  if (tid < 30 && (tid < 14 || tid >= 16)) {
    const bool  isY  = tid < GH_;
    const int   g    = isY ? tid : tid - 16;
    const float strt = isY ? ry1 : rx1;
    const float step = isY ? step_y : step_x;
    const float gpos = strt + ((float)g + 0.5f) * step;
    const float v    = (gpos >= -1.0f && gpos <= 100.0f) ? 1.0f : 0.0f;
    const float p    = fminf(fmaxf(gpos, 0.0f), 99.0f);
    const int   p0   = (int)floorf(p);
    const float l    = p - (float)p0;
    const int   p1   = min(p0 + 1, 99);
    if (isY) { iy0[g] = p0; iy1[g] = p1; fly[g] = l; fvy[g] = v; }
    else     { ix0[g] = p0; ix1[g] = p1; flx[g] = l; fvx[g] = v; }
  }
  __syncthreads();

  const size_t base = (size_t)b * (size_t)bStride + (size_t)tid * (size_t)cStride;

  for (int bin = 0; bin < NBIN_; ++bin) {
    const int ph = bin / PW_;
    const int pw = bin - ph * PW_;
    float acc = 0.0f;
#pragma unroll
    for (int sy = 0; sy < SR_; ++sy) {
      const int   gi = ph * SR_ + sy;
      const int   y0 = iy0[gi], y1 = iy1[gi];
      const float ly = fly[gi], hy = 1.0f - ly, vy = fvy[gi];
      const size_t o0 = base + (size_t)y0 * (size_t)yStride;
      const size_t o1 = base + (size_t)y1 * (size_t)yStride;
#pragma unroll
      for (int sx = 0; sx < SR_; ++sx) {
        const int   gj = pw * SR_ + sx;
        const int   x0 = ix0[gj], x1 = ix1[gj];
        const float lx = flx[gj], vxy = fvx[gj] * vy;
        const float hx = 1.0f - lx;
        const float f00 = feat[o0 + (size_t)x0 * (size_t)xStride];
        const float f01 = feat[o0 + (size_t)x1 * (size_t)xStride];
        const float f10 = feat[o1 + (size_t)x0 * (size_t)xStride];
        const float f11 = feat[o1 + (size_t)x1 * (size_t)xStride];
        const float val = hy * (hx * f00 + lx * f01) + ly * (hx * f10 + lx * f11);
        acc += val * vxy;
      }
    }
    sout[tid * NBIN_ + bin] = acc * 0.25f;           // mean over SR*SR samples
  }
  __syncthreads();

  // out[k][c][ph][pw] flat == k*12544 + (c*49 + bin): stream LDS out as float4
  // (3136 b128 stores per ROI instead of 12544 b32 stores on the 50 MB
  //  HBM-write stream). 12544 % 4 == 0, base is 16B-aligned.
  float4* __restrict__ out4 = (float4*)(out + (size_t)k * OUTPERROI_);
  const float4* __restrict__ s4 = (const float4*)sout;
  for (int i = tid; i < OUTPERROI_ / 4; i += 256) {
    out4[i] = s4[i];
  }
}

extern "C" void kernel_launch(void* const* d_in, const int* in_sizes, int n_in,
                              void* d_out, int out_size, void* d_ws, size_t ws_size,
                              hipStream_t stream) {
  const float* feat = (const float*)d_in[0];
  const float* rois = (const float*)d_in[1];
  float* out = (float*)d_out;

  const int B = in_sizes[0] / (C_ * HW_);            // = 2
  const int K = in_sizes[1] / 5;                     // = 1000

  const size_t nhwc_bytes = (size_t)B * HW_ * C_ * sizeof(float);
  if (ws_size >= nhwc_bytes) {
    // Stage 1: transpose to NHWC (async global->LDS path), then coalesced gather.
    float* nhwc = (float*)d_ws;
    dim3 tb(32, 8, 1);
    dim3 tg((HW_ + 31) / 32, C_ / 32, B);
    nchw_to_nhwc_async<<<tg, tb, 0, stream>>>(feat, nhwc);
    roi_align_kernel<<<K, 256, 0, stream>>>(
        nhwc, rois, out,
        (long)HW_ * C_, /*yStride=*/W_ * C_, /*xStride=*/C_, /*cStride=*/1);
  } else {
    // Fallback: gather straight from NCHW.
    roi_align_kernel<<<K, 256, 0, stream>>>(
        feat, rois, out,
        (long)C_ * HW_, /*yStride=*/W_, /*xStride=*/1, /*cStride=*/HW_);
  }
}